// DBPNet_1769526526344
// MI455X (gfx1250) — compile-verified
//
#include <hip/hip_runtime.h>
#include <math.h>

// ----------------------------------------------------------------------------
// DBP-Net forward for MI455X (gfx1250).
// All GEMM-shaped work (AAH, A^H y, ADMM einsums, Newton-Schulz matrix
// inversion) runs through V_WMMA_F32_16X16X4_F32 (full f32: the ADMM fixed
// point + matrix inverse are precision sensitive). Working set is L2-resident
// (Ac = 8MB << 192MB L2), so the problem is matrix-pipe bound, not HBM bound.
// inv(AAH + rho I) is computed with Newton-Schulz X' = 2X - X*M*X seeded by
// X0 = I/||M||inf (M Hermitian PD => guaranteed convergence), i.e. the inverse
// is itself just 32 more 512^3 complex WMMA GEMMs per outer iteration.
// ----------------------------------------------------------------------------

typedef __attribute__((ext_vector_type(2))) float v2f;
typedef __attribute__((ext_vector_type(8))) float v8f;

__device__ __forceinline__ v8f wmma_f32x4(v2f a, v2f b, v8f c) {
  // D = A(16x4,f32) * B(4x16,f32) + C(16x16,f32); 8-arg form:
  // (neg_a, A, neg_b, B, c_mod, C, reuse_a, reuse_b)
  return __builtin_amdgcn_wmma_f32_16x16x4_f32(false, a, false, b, (short)0, c,
                                               false, false);
}

// ---------------------------------------------------------------------------
// Generic complex GEMM:  C = beta*E + alpha * (A x opB(B))
//   A: [M][lda] row-major complex (always normal)
//   B: transB ? [N][ldb] : [K][ldb]  row-major complex; conjB conjugates B
//   E: optional (nullptr => beta term skipped); scalars from device ptr `scal`
//      (alpha=scal[scalIdx], beta=scal[scalIdx+1]) if scal!=nullptr.
// Requires M%64==0, N%128==0, K%32==0 (true for all call sites here).
// ---------------------------------------------------------------------------
#define BM 64
#define BN 128
#define BK 32
#define LDS_K (BK + 2)

__launch_bounds__(256, 2)
__global__ void cgemm_wmma_f32(const float2* __restrict__ A, int lda,
                               const float2* __restrict__ B, int ldb,
                               const float2* __restrict__ E, int lde,
                               float2* __restrict__ C, int ldc,
                               int M, int N, int K,
                               int transB, int conjB,
                               float alphaI, float betaI,
                               const float* __restrict__ scal, int scalIdx) {
  __shared__ float As_r[BM][LDS_K];
  __shared__ float As_i[BM][LDS_K];
  __shared__ float Bs_r[BN][LDS_K];   // stored [n][k]
  __shared__ float Bs_i[BN][LDS_K];

  const int tid   = threadIdx.x;
  const int lane  = tid & 31;
  const int wave  = tid >> 5;     // 0..7
  const int waveM = wave >> 2;    // 0..1
  const int waveN = wave & 3;     // 0..3

  const int bM = blockIdx.y * BM;
  const int bN = blockIdx.x * BN;

  float alpha = alphaI, beta = betaI;
  if (scal) { alpha = scal[scalIdx]; beta = scal[scalIdx + 1]; }
  const float bsgn = conjB ? -1.0f : 1.0f;

  v8f cr[2][2], ci[2][2];
  #pragma unroll
  for (int mi = 0; mi < 2; ++mi)
    #pragma unroll
    for (int ni = 0; ni < 2; ++ni)
      #pragma unroll
      for (int q = 0; q < 8; ++q) { cr[mi][ni][q] = 0.0f; ci[mi][ni][q] = 0.0f; }

  const int mrow = lane & 15;
  const int koff = (lane >> 4) * 2;

  for (int k0 = 0; k0 < K; k0 += BK) {
    // ---- stage A tile [64 x 32] ----
    {
      const int m  = tid >> 2;
      const int kq = (tid & 3) * 8;
      const float2* ap = A + (size_t)(bM + m) * lda + (k0 + kq);
      if (k0 + BK < K) __builtin_prefetch((const void*)(ap + BK), 0, 1);
      #pragma unroll
      for (int j = 0; j < 8; ++j) {
        float2 v = ap[j];
        As_r[m][kq + j] = v.x;
        As_i[m][kq + j] = v.y;
      }
    }
    // ---- stage B tile into Bs[n][k] ----
    if (transB) {           // mem [N][ldb]
      #pragma unroll
      for (int rep = 0; rep < 2; ++rep) {
        const int n  = (tid >> 2) + rep * 64;
        const int kq = (tid & 3) * 8;
        const float2* bp = B + (size_t)(bN + n) * ldb + (k0 + kq);
        #pragma unroll
        for (int j = 0; j < 8; ++j) {
          float2 v = bp[j];
          Bs_r[n][kq + j] = v.x;
          Bs_i[n][kq + j] = v.y * bsgn;
        }
      }
    } else {                // mem [K][ldb]
      const int kRow = tid >> 5;          // 0..7
      const int nq   = (tid & 31) * 4;    // 0..124
      #pragma unroll
      for (int kk = 0; kk < 4; ++kk) {
        const int k = kk * 8 + kRow;
        const float2* bp = B + (size_t)(k0 + k) * ldb + (bN + nq);
        #pragma unroll
        for (int j = 0; j < 4; ++j) {
          float2 v = bp[j];
          Bs_r[nq + j][k] = v.x;
          Bs_i[nq + j][k] = v.y * bsgn;
        }
      }
    }
    __syncthreads();

    // ---- 8 k-steps of 16x16x4 WMMA ----
    #pragma unroll
    for (int ks = 0; ks < BK / 4; ++ks) {
      const int kb = ks * 4 + koff;
      v2f ar[2], ai[2], ain[2], br[2], bi[2];
      #pragma unroll
      for (int mi = 0; mi < 2; ++mi) {
        const int r = waveM * 32 + mi * 16 + mrow;
        ar[mi][0] = As_r[r][kb];  ar[mi][1] = As_r[r][kb + 1];
        ai[mi][0] = As_i[r][kb];  ai[mi][1] = As_i[r][kb + 1];
        ain[mi] = -ai[mi];   // f32 WMMA has no A-neg modifier: pre-negate
      }
      #pragma unroll
      for (int ni = 0; ni < 2; ++ni) {
        const int c = waveN * 32 + ni * 16 + mrow;
        br[ni][0] = Bs_r[c][kb];  br[ni][1] = Bs_r[c][kb + 1];
        bi[ni][0] = Bs_i[c][kb];  bi[ni][1] = Bs_i[c][kb + 1];
      }
      #pragma unroll
      for (int mi = 0; mi < 2; ++mi)
        #pragma unroll
        for (int ni = 0; ni < 2; ++ni) {
          cr[mi][ni] = wmma_f32x4(ar[mi],  br[ni], cr[mi][ni]);
          cr[mi][ni] = wmma_f32x4(ain[mi], bi[ni], cr[mi][ni]);
          ci[mi][ni] = wmma_f32x4(ar[mi],  bi[ni], ci[mi][ni]);
          ci[mi][ni] = wmma_f32x4(ai[mi],  br[ni], ci[mi][ni]);
        }
    }
    __syncthreads();
  }

  // ---- epilogue: C = beta*E + alpha*acc ----
  // 16x16 f32 D layout: vgpr j, lanes0-15 -> M=j ; lanes16-31 -> M=j+8
  const int lcol = lane & 15;
  const int rAdd = (lane >> 4) * 8;
  #pragma unroll
  for (int mi = 0; mi < 2; ++mi) {
    #pragma unroll
    for (int ni = 0; ni < 2; ++ni) {
      const int col = bN + waveN * 32 + ni * 16 + lcol;
      #pragma unroll
      for (int j = 0; j < 8; ++j) {
        const int m = bM + waveM * 32 + mi * 16 + rAdd + j;
        float vr = alpha * cr[mi][ni][j];
        float vi = alpha * ci[mi][ni][j];
        if (E) {
          float2 e = E[(size_t)m * lde + col];
          vr += beta * e.x;
          vi += beta * e.y;
        }
        C[(size_t)m * ldc + col] = make_float2(vr, vi);
      }
    }
  }
  (void)M; (void)N;
}

// ---------------------------------------------------------------------------
// Elementwise / pack / reduction kernels
// ---------------------------------------------------------------------------
__global__ void pack_ac_kernel(const float* __restrict__ A, float2* __restrict__ Ac,
                               int Nv, int Nt) {
  size_t i = (size_t)blockIdx.x * blockDim.x + threadIdx.x;
  size_t n = (size_t)Nv * Nt;
  if (i < n) Ac[i] = make_float2(A[i], A[n + i]);
}

__global__ void pack_yc_kernel(const float* __restrict__ y, float2* __restrict__ yc,
                               int B, int Nv) {
  size_t i = (size_t)blockIdx.x * blockDim.x + threadIdx.x;
  if (i < (size_t)B * Nv) {
    size_t b = i / Nv, v = i % Nv;
    const float* yb = y + b * 2 * Nv;
    yc[i] = make_float2(yb[v], yb[Nv + v]);
  }
}

__global__ void zero2_kernel(float2* __restrict__ p, size_t n) {
  size_t i = (size_t)blockIdx.x * blockDim.x + threadIdx.x;
  if (i < n) p[i] = make_float2(0.0f, 0.0f);
}

__global__ void copy2_kernel(const float2* __restrict__ s, float2* __restrict__ d, size_t n) {
  size_t i = (size_t)blockIdx.x * blockDim.x + threadIdx.x;
  if (i < n) d[i] = s[i];
}

__global__ void sub2_kernel(const float2* __restrict__ a, const float2* __restrict__ b,
                            float2* __restrict__ d, size_t n) {
  size_t i = (size_t)blockIdx.x * blockDim.x + threadIdx.x;
  if (i < n) {
    float2 x = a[i], y = b[i];
    d[i] = make_float2(x.x - y.x, x.y - y.y);
  }
}

// scal: [0]=rho [1]=eps [2]=rho/(rho+1e-8) [3]=1/(rho+1e-8) [4]=1/||M||inf
__global__ void iter_scalars_kernel(const float* __restrict__ log_rho,
                                    const float* __restrict__ log_eps, int it,
                                    float* __restrict__ s, unsigned* __restrict__ cell) {
  float rho = __expf(log_rho[it]);
  float eps = __expf(log_eps[it]);
  s[0] = rho;
  s[1] = eps;
  s[2] = rho / (rho + 1e-8f);
  s[3] = 1.0f / (rho + 1e-8f);
  *cell = 0u;
}

__global__ void build_m_kernel(const float2* __restrict__ AAH, const float* __restrict__ s,
                               float2* __restrict__ Mm, int n) {
  size_t i = (size_t)blockIdx.x * blockDim.x + threadIdx.x;
  if (i < (size_t)n * n) {
    float2 v = AAH[i];
    if ((i / n) == (i % n)) v.x += s[0];
    Mm[i] = v;
  }
}

__global__ void rownorm_kernel(const float2* __restrict__ Mm, int n,
                               unsigned* __restrict__ cell) {
  const int row = blockIdx.x;
  const int tid = threadIdx.x;
  float s = 0.0f;
  const float2* p = Mm + (size_t)row * n;
  for (int c = tid; c < n; c += blockDim.x) {
    float2 v = p[c];
    s += fabsf(v.x) + fabsf(v.y);   // |z| <= |re|+|im|: safe upper bound
  }
  __shared__ float red[256];
  red[tid] = s;
  __syncthreads();
  for (int st = 128; st > 0; st >>= 1) {
    if (tid < st) red[tid] += red[tid + st];
    __syncthreads();
  }
  if (tid == 0) atomicMax(cell, __float_as_uint(red[0]));  // values >= 0
}

__global__ void finish_norm_kernel(const unsigned* __restrict__ cell,
                                   float* __restrict__ s) {
  float m = __uint_as_float(*cell);
  s[4] = 1.0f / fmaxf(m, 1e-20f);
}

__global__ void init_x_kernel(float2* __restrict__ X, const float* __restrict__ s, int n) {
  size_t i = (size_t)blockIdx.x * blockDim.x + threadIdx.x;
  if (i < (size_t)n * n) {
    float d = ((i / n) == (i % n)) ? s[4] : 0.0f;
    X[i] = make_float2(d, 0.0f);
  }
}

// ---- denoiser ----
__global__ void conv1_kernel(const float2* __restrict__ x, const float* __restrict__ w,
                             float* __restrict__ out, int B, int L) {
  const int l = blockIdx.x * blockDim.x + threadIdx.x;
  const int f = blockIdx.y;
  const int b = blockIdx.z;
  if (l >= L) return;
  const float2* xb = x + (size_t)b * L;
  float acc = 0.0f;
  #pragma unroll
  for (int dk = 0; dk < 3; ++dk) {
    const int ll = l + dk - 1;
    if (ll < 0 || ll >= L) continue;
    float2 v = xb[ll];
    acc += w[f * 6 + dk] * v.x + w[f * 6 + 3 + dk] * v.y;
  }
  out[((size_t)b * 32 + f) * L + l] = acc;
}

__global__ void bn_stats_kernel(const float* __restrict__ t, float* __restrict__ mv,
                                int B, int C, int L) {
  const int f = blockIdx.x;
  const int tid = threadIdx.x;
  float s = 0.0f, s2 = 0.0f;
  for (int b = 0; b < B; ++b) {
    const float* p = t + ((size_t)b * C + f) * L;
    for (int l = tid; l < L; l += blockDim.x) {
      float v = p[l];
      s += v;
      s2 += v * v;
    }
  }
  __shared__ float r1[256], r2[256];
  r1[tid] = s; r2[tid] = s2;
  __syncthreads();
  for (int st = 128; st > 0; st >>= 1) {
    if (tid < st) { r1[tid] += r1[tid + st]; r2[tid] += r2[tid + st]; }
    __syncthreads();
  }
  if (tid == 0) {
    float inv = 1.0f / ((float)B * (float)L);
    float m = r1[0] * inv;
    mv[f * 2] = m;
    mv[f * 2 + 1] = r2[0] * inv - m * m;   // biased var (matches jnp.var)
  }
}

__global__ void bn_relu_kernel(float* __restrict__ t, const float* __restrict__ mv,
                               const float* __restrict__ g, const float* __restrict__ bb,
                               int C, int L, size_t total) {
  size_t i = (size_t)blockIdx.x * blockDim.x + threadIdx.x;
  if (i >= total) return;
  int f = (int)((i / (size_t)L) % (size_t)C);
  float m = mv[f * 2], v = mv[f * 2 + 1];
  float r = g[f] * (t[i] - m) * rsqrtf(v + 1e-5f) + bb[f];
  t[i] = fmaxf(r, 0.0f);
}

__global__ void conv2_kernel(const float* __restrict__ in, const float* __restrict__ w,
                             float* __restrict__ out, int B, int L) {
  __shared__ float wsh[32 * 32 * 3];
  for (int i = threadIdx.x; i < 32 * 32 * 3; i += blockDim.x) wsh[i] = w[i];
  __syncthreads();
  const int l = blockIdx.x * blockDim.x + threadIdx.x;
  const int f = blockIdx.y;
  const int b = blockIdx.z;
  const float* wf = wsh + f * 96;
  float acc = 0.0f;
  #pragma unroll 4
  for (int c = 0; c < 32; ++c) {
    const float* p = in + ((size_t)b * 32 + c) * L;
    float x0 = (l > 0) ? p[l - 1] : 0.0f;
    float x1 = p[l];
    float x2 = (l < L - 1) ? p[l + 1] : 0.0f;
    acc += wf[c * 3] * x0 + wf[c * 3 + 1] * x1 + wf[c * 3 + 2] * x2;
  }
  out[((size_t)b * 32 + f) * L + l] = acc;
}

__global__ void convf_res_kernel(const float* __restrict__ in, const float* __restrict__ w,
                                 const float* __restrict__ bias,
                                 const float2* __restrict__ x, float2* __restrict__ rc,
                                 int B, int L) {
  __shared__ float wsh[2 * 32 * 3];
  for (int i = threadIdx.x; i < 2 * 32 * 3; i += blockDim.x) wsh[i] = w[i];
  __syncthreads();
  const int l = blockIdx.x * blockDim.x + threadIdx.x;
  const int b = blockIdx.z;
  float a0 = bias[0], a1 = bias[1];
  #pragma unroll 4
  for (int c = 0; c < 32; ++c) {
    const float* p = in + ((size_t)b * 32 + c) * L;
    float x0 = (l > 0) ? p[l - 1] : 0.0f;
    float x1 = p[l];
    float x2 = (l < L - 1) ? p[l + 1] : 0.0f;
    a0 += wsh[c * 3] * x0 + wsh[c * 3 + 1] * x1 + wsh[c * 3 + 2] * x2;
    a1 += wsh[96 + c * 3] * x0 + wsh[96 + c * 3 + 1] * x1 + wsh[96 + c * 3 + 2] * x2;
  }
  float2 xv = x[(size_t)b * L + l];
  rc[(size_t)b * L + l] = make_float2(xv.x + a0, xv.y + a1);
}

// ---- ADMM projection: v = Ax+u ; z = yc + min(1, eps/||v-yc||)*(v-yc) ; u += Ax - z
__global__ void admm_proj_kernel(const float2* __restrict__ yc, const float2* __restrict__ Ax,
                                 float2* __restrict__ z, float2* __restrict__ u,
                                 const float* __restrict__ s, int Nv) {
  const int b = blockIdx.x;
  const int tid = threadIdx.x;
  const float2* ycb = yc + (size_t)b * Nv;
  const float2* axb = Ax + (size_t)b * Nv;
  float2* zb = z + (size_t)b * Nv;
  float2* ub = u + (size_t)b * Nv;

  float local = 0.0f;
  for (int v = tid; v < Nv; v += blockDim.x) {
    float2 uv = ub[v], av = axb[v], yv = ycb[v];
    float dr = av.x + uv.x - yv.x;
    float di = av.y + uv.y - yv.y;
    local += dr * dr + di * di;
  }
  __shared__ float red[256];
  __shared__ float fac;
  red[tid] = local;
  __syncthreads();
  for (int st = 128; st > 0; st >>= 1) {
    if (tid < st) red[tid] += red[tid + st];
    __syncthreads();
  }
  if (tid == 0) {
    float nrm = sqrtf(red[0]);
    fac = fminf(1.0f, s[1] / (nrm + 1e-12f));
  }
  __syncthreads();
  const float f = fac;
  for (int v = tid; v < Nv; v += blockDim.x) {
    float2 uv = ub[v], av = axb[v], yv = ycb[v];
    float dr = av.x + uv.x - yv.x;
    float di = av.y + uv.y - yv.y;
    float zr = yv.x + f * dr;
    float zi = yv.y + f * di;
    zb[v] = make_float2(zr, zi);
    ub[v] = make_float2(uv.x + av.x - zr, uv.y + av.y - zi);
  }
}

__global__ void unpack_out_kernel(const float2* __restrict__ x, float* __restrict__ out,
                                  int B, int Nt) {
  size_t i = (size_t)blockIdx.x * blockDim.x + threadIdx.x;
  if (i < (size_t)B * Nt) {
    size_t b = i / Nt, t = i % Nt;
    float2 v = x[i];
    out[b * 2 * Nt + t] = v.x;
    out[b * 2 * Nt + Nt + t] = v.y;
  }
}

// ---------------------------------------------------------------------------
// Host orchestration
// ---------------------------------------------------------------------------
static inline void launch_cgemm(hipStream_t st,
                                const float2* A, int lda, const float2* B, int ldb,
                                const float2* E, int lde, float2* C, int ldc,
                                int M, int N, int K, int transB, int conjB,
                                float alpha, float beta,
                                const float* scal, int scalIdx) {
  dim3 g(N / BN, M / BM), blk(256);
  cgemm_wmma_f32<<<g, blk, 0, st>>>(A, lda, B, ldb, E, lde, C, ldc, M, N, K,
                                    transB, conjB, alpha, beta, scal, scalIdx);
}

extern "C" void kernel_launch(void* const* d_in, const int* in_sizes, int n_in,
                              void* d_out, int out_size, void* d_ws, size_t ws_size,
                              hipStream_t stream) {
  (void)in_sizes; (void)n_in; (void)out_size; (void)ws_size;
  const int B = 128, Nv = 512, Nt = 2048, F = 32;
  const int ITERS = 5, ADMM = 3, NSI = 16;

  const float* y_in = (const float*)d_in[0];
  const float* A_in = (const float*)d_in[1];
  const float* w1   = (const float*)d_in[2];
  const float* g1   = (const float*)d_in[3];
  const float* b1   = (const float*)d_in[4];
  const float* w2   = (const float*)d_in[5];
  const float* g2   = (const float*)d_in[6];
  const float* b2   = (const float*)d_in[7];
  const float* wf   = (const float*)d_in[8];
  const float* bf   = (const float*)d_in[9];
  const float* lrho = (const float*)d_in[10];
  const float* leps = (const float*)d_in[11];

  char* wsb = (char*)d_ws;
  size_t off = 0;
  auto alloc = [&](size_t bytes) -> void* {
    void* p = (void*)(wsb + off);
    off += (bytes + 255) & ~(size_t)255;
    return p;
  };
  float2* Ac   = (float2*)alloc((size_t)Nv * Nt * sizeof(float2));
  float2* yc   = (float2*)alloc((size_t)B * Nv * sizeof(float2));
  float2* AAH  = (float2*)alloc((size_t)Nv * Nv * sizeof(float2));
  float2* Mm   = (float2*)alloc((size_t)Nv * Nv * sizeof(float2));
  float2* Xa   = (float2*)alloc((size_t)Nv * Nv * sizeof(float2));
  float2* Xb   = (float2*)alloc((size_t)Nv * Nv * sizeof(float2));
  float2* T    = (float2*)alloc((size_t)Nv * Nv * sizeof(float2));
  float2* x    = (float2*)alloc((size_t)B * Nt * sizeof(float2));
  float2* rc   = (float2*)alloc((size_t)B * Nt * sizeof(float2));
  float2* tpx  = (float2*)alloc((size_t)B * Nt * sizeof(float2));
  float2* u    = (float2*)alloc((size_t)B * Nv * sizeof(float2));
  float2* z    = (float2*)alloc((size_t)B * Nv * sizeof(float2));
  float2* zmu  = (float2*)alloc((size_t)B * Nv * sizeof(float2));
  float2* Atx  = (float2*)alloc((size_t)B * Nv * sizeof(float2));
  float2* tmv  = (float2*)alloc((size_t)B * Nv * sizeof(float2));
  float2* Ax   = (float2*)alloc((size_t)B * Nv * sizeof(float2));
  float*  tmp1 = (float*)alloc((size_t)B * F * Nt * sizeof(float));
  float*  tmp2 = (float*)alloc((size_t)B * F * Nt * sizeof(float));
  float*  mv1  = (float*)alloc(2 * F * sizeof(float));
  float*  mv2  = (float*)alloc(2 * F * sizeof(float));
  float*  scal = (float*)alloc(16 * sizeof(float));
  unsigned* cell = (unsigned*)alloc(64);

  const size_t nAc = (size_t)Nv * Nt;
  const size_t nYc = (size_t)B * Nv;
  const size_t nX  = (size_t)B * Nt;
  const size_t nM2 = (size_t)Nv * Nv;

  // pack complex operands
  pack_ac_kernel<<<(nAc + 255) / 256, 256, 0, stream>>>(A_in, Ac, Nv, Nt);
  pack_yc_kernel<<<(nYc + 255) / 256, 256, 0, stream>>>(y_in, yc, B, Nv);

  // AAH[v][w] = sum_t Ac[v][t] * conj(Ac[w][t])   (B operand: [N][K] mem, conj)
  launch_cgemm(stream, Ac, Nt, Ac, Nt, nullptr, 0, AAH, Nv,
               Nv, Nv, Nt, /*transB=*/1, /*conjB=*/1, 1.0f, 0.0f, nullptr, 0);
  // x0 = yc @ conj(Ac)          ([B,Nv] x [Nv,Nt])
  launch_cgemm(stream, yc, Nv, Ac, Nt, nullptr, 0, x, Nt,
               B, Nt, Nv, /*transB=*/0, /*conjB=*/1, 1.0f, 0.0f, nullptr, 0);
  // u = 0
  zero2_kernel<<<(nYc + 255) / 256, 256, 0, stream>>>(u, nYc);

  const dim3 convGrid(Nt / 256, F, B), convBlk(256);
  const size_t actTotal = (size_t)B * F * Nt;

  for (int it = 0; it < ITERS; ++it) {
    iter_scalars_kernel<<<1, 1, 0, stream>>>(lrho, leps, it, scal, cell);

    // ---------------- denoiser: rc = x + CNN(x) ----------------
    conv1_kernel<<<convGrid, convBlk, 0, stream>>>(x, w1, tmp1, B, Nt);
    bn_stats_kernel<<<F, 256, 0, stream>>>(tmp1, mv1, B, F, Nt);
    bn_relu_kernel<<<(actTotal + 255) / 256, 256, 0, stream>>>(tmp1, mv1, g1, b1, F, Nt, actTotal);
    conv2_kernel<<<convGrid, convBlk, 0, stream>>>(tmp1, w2, tmp2, B, Nt);
    bn_stats_kernel<<<F, 256, 0, stream>>>(tmp2, mv2, B, F, Nt);
    bn_relu_kernel<<<(actTotal + 255) / 256, 256, 0, stream>>>(tmp2, mv2, g2, b2, F, Nt, actTotal);
    convf_res_kernel<<<dim3(Nt / 256, 1, B), convBlk, 0, stream>>>(tmp2, wf, bf, x, rc, B, Nt);

    // ---------------- M_inv via Newton-Schulz ----------------
    build_m_kernel<<<(nM2 + 255) / 256, 256, 0, stream>>>(AAH, scal, Mm, Nv);
    rownorm_kernel<<<Nv, 256, 0, stream>>>(Mm, Nv, cell);
    finish_norm_kernel<<<1, 1, 0, stream>>>(cell, scal);
    init_x_kernel<<<(nM2 + 255) / 256, 256, 0, stream>>>(Xa, scal, Nv);
    float2* Xc = Xa;
    float2* Xn = Xb;
    for (int k = 0; k < NSI; ++k) {
      // T = M * Xc
      launch_cgemm(stream, Mm, Nv, Xc, Nv, nullptr, 0, T, Nv,
                   Nv, Nv, Nv, 0, 0, 1.0f, 0.0f, nullptr, 0);
      // Xn = 2*Xc - Xc*T
      launch_cgemm(stream, Xc, Nv, T, Nv, Xc, Nv, Xn, Nv,
                   Nv, Nv, Nv, 0, 0, -1.0f, 2.0f, nullptr, 0);
      float2* t2 = Xc; Xc = Xn; Xn = t2;
    }
    const float2* Minv = Xc;

    // ---------------- inner ADMM ----------------
    copy2_kernel<<<(nYc + 255) / 256, 256, 0, stream>>>(yc, z, nYc);
    for (int j = 0; j < ADMM; ++j) {
      sub2_kernel<<<(nYc + 255) / 256, 256, 0, stream>>>(z, u, zmu, nYc);
      // temp_x = rc/(rho+1e-8) + (rho/(rho+1e-8)) * (zmu @ conj(Ac))
      launch_cgemm(stream, zmu, Nv, Ac, Nt, rc, Nt, tpx, Nt,
                   B, Nt, Nv, 0, 1, 1.0f, 1.0f, scal, 2);
      // Atx = temp_x @ Ac^T
      launch_cgemm(stream, tpx, Nt, Ac, Nt, nullptr, 0, Atx, Nv,
                   B, Nv, Nt, 1, 0, 1.0f, 0.0f, nullptr, 0);
      // tmv = Atx @ Minv^T
      launch_cgemm(stream, Atx, Nv, Minv, Nv, nullptr, 0, tmv, Nv,
                   B, Nv, Nv, 1, 0, 1.0f, 0.0f, nullptr, 0);
      // x = temp_x - tmv @ conj(Ac)
      launch_cgemm(stream, tmv, Nv, Ac, Nt, tpx, Nt, x, Nt,
                   B, Nt, Nv, 0, 1, -1.0f, 1.0f, nullptr, 0);
      // Ax = x @ Ac^T
      launch_cgemm(stream, x, Nt, Ac, Nt, nullptr, 0, Ax, Nv,
                   B, Nv, Nt, 1, 0, 1.0f, 0.0f, nullptr, 0);
      // projection + dual update
      admm_proj_kernel<<<B, 256, 0, stream>>>(yc, Ax, z, u, scal, Nv);
    }
  }

  unpack_out_kernel<<<(nX + 255) / 256, 256, 0, stream>>>(x, (float*)d_out, B, Nt);
}